// DirectedGraphBuilder_36094905155926
// MI455X (gfx1250) — compile-verified
//
#include <hip/hip_runtime.h>
#include <stdint.h>

// DirectedGraphBuilder for MI455X (gfx1250, wave32).
// One wave32 per row of [V, 64]; lane l owns columns 2l, 2l+1.
// nidx rows are staged into LDS via double-buffered
// global_load_async_to_lds_b64 (ASYNCcnt pipeline), score gathers hit the
// L2-resident 4 MB table, the drop/keep decision is forced scalar
// (s_cbranch) so dropped rows issue no dist loads at all, and all streamed
// dist/nidx traffic uses non-temporal hints to keep score L2-resident.

#define KDIM 64
#define WAVES_PER_BLOCK 8
#define BLOCK_THREADS (WAVES_PER_BLOCK * 32)
#define DEFAULT_SCORE (-10000.0f)

typedef __attribute__((ext_vector_type(2))) float v2f;
typedef __attribute__((ext_vector_type(2))) int v2i;

__global__ __launch_bounds__(BLOCK_THREADS) void
DirectedGraphBuilder_gfx1250_kernel(const float* __restrict__ dist,
                                    const int* __restrict__ nidx,
                                    const float* __restrict__ score,
                                    float* __restrict__ dist_out,
                                    int* __restrict__ nidx_out,
                                    int V)
{
    // 8 waves x 2 buffers x 64 ints = 4 KB LDS staging for nidx rows.
    __shared__ int lds_rows[WAVES_PER_BLOCK][2][KDIM];

    const int lane  = threadIdx.x & 31;
    const int wave  = threadIdx.x >> 5;
    const int gwave = (int)((blockIdx.x * blockDim.x + threadIdx.x) >> 5);
    const int nwave = (int)((gridDim.x * blockDim.x) >> 5);

    // LDS byte offsets of this wave's two staging buffers. Low 32 bits of a
    // flat pointer to a __shared__ object are the LDS byte offset (ISA 10.2).
    uint32_t lds_off0 = (uint32_t)(uintptr_t)&lds_rows[wave][0][0];
    uint32_t lds_off1 = (uint32_t)(uintptr_t)&lds_rows[wave][1][0];

    int row = gwave;
    if (row >= V) return;

    // Issue one async 8B-per-lane copy of nidx row r into an LDS buffer.
    // One instruction per wave -> ASYNCcnt += 1 per staged row.
    auto issue_row = [&](int r, uint32_t lds_base) {
        uint64_t g = (uint64_t)(uintptr_t)(nidx + (size_t)r * KDIM + (size_t)(lane * 2));
        uint32_t l = lds_base + (uint32_t)(lane * 8);
        asm volatile("global_load_async_to_lds_b64 %0, %1, off"
                     :: "v"(l), "v"(g)
                     : "memory");
    };

    issue_row(row, lds_off0);
    int buf = 0;

    for (int r = row; r < V; r += nwave) {
        int next = r + nwave;   // wave-uniform
        if (next < V) {
            // Prefetch next row into the other buffer, then wait until only
            // that prefetch is outstanding: async loads complete in order,
            // so ASYNCcnt <= 1 means the current buffer is fully in LDS.
            issue_row(next, buf ? lds_off0 : lds_off1);
            asm volatile("s_wait_asynccnt 0x1" ::: "memory");
        } else {
            asm volatile("s_wait_asynccnt 0x0" ::: "memory");
        }

        // Lane l owns row elements 2l, 2l+1 (ds_load_b64).
        int i0 = lds_rows[wave][buf][2 * lane];
        int i1 = lds_rows[wave][buf][2 * lane + 1];

        // Gather neighbour scores from the L2-resident 4 MB score table.
        float s0 = DEFAULT_SCORE, s1 = DEFAULT_SCORE;
        if (i0 >= 0) s0 = score[i0];
        if (i1 >= 0) s1 = score[i1];
        if (lane == 0) s0 = DEFAULT_SCORE;   // column 0 = self, excluded from max

        // Wave32 max reduction (5 steps) -> m identical in all lanes.
        float m = fmaxf(s0, s1);
        #pragma unroll
        for (int off = 16; off > 0; off >>= 1)
            m = fmaxf(m, __shfl_xor(m, off, 32));

        // r is wave-uniform: pin it to an SGPR so the self-score read becomes
        // a scalar (KMcnt) load and the drop branch a scalar s_cbranch.
        int ru = __builtin_amdgcn_readfirstlane(r);
        float self_score = score[ru];
        int drop = __builtin_amdgcn_readfirstlane(self_score < m ? 1 : 0);

        size_t base = (size_t)r * KDIM + (size_t)(lane * 2);
        if (drop) {
            // ~98% of rows with i.i.d. scores: no dist read issued at all.
            v2f z = {0.0f, 0.0f};
            __builtin_nontemporal_store(z, (v2f*)(dist_out + base));
            v2i nv;
            nv.x = (lane == 0) ? i0 : -1;    // keep self index, drop neighbours
            nv.y = -1;
            __builtin_nontemporal_store(nv, (v2i*)(nidx_out + base));
        } else {
            v2f dv = __builtin_nontemporal_load((const v2f*)(dist + base));
            __builtin_nontemporal_store(dv, (v2f*)(dist_out + base));
            v2i nv = {i0, i1};
            __builtin_nontemporal_store(nv, (v2i*)(nidx_out + base));
        }

        buf ^= 1;
    }
}

extern "C" void kernel_launch(void* const* d_in, const int* in_sizes, int n_in,
                              void* d_out, int out_size, void* d_ws, size_t ws_size,
                              hipStream_t stream) {
    (void)n_in; (void)out_size; (void)d_ws; (void)ws_size;

    const float* dist  = (const float*)d_in[0];   // [V, 64] f32
    const int*   nidx  = (const int*)  d_in[1];   // [V, 64] i32
    const float* score = (const float*)d_in[2];   // [V, 1]  f32

    const int V = in_sizes[2];                    // score has V elements

    float* dist_out = (float*)d_out;                            // [V, 64] f32
    int*   nidx_out = (int*)((float*)d_out + (size_t)V * KDIM); // [V, 64] i32 bits

    // Persistent-ish grid: enough wave32s to saturate 23.3 TB/s HBM, each
    // wave grid-strides over rows with the double-buffered async pipeline.
    int blocks = 3072;
    int max_blocks = (V + WAVES_PER_BLOCK - 1) / WAVES_PER_BLOCK;
    if (blocks > max_blocks) blocks = max_blocks;
    if (blocks < 1) blocks = 1;

    DirectedGraphBuilder_gfx1250_kernel<<<blocks, BLOCK_THREADS, 0, stream>>>(
        dist, nidx, score, dist_out, nidx_out, V);
}